// QKRotationWrapper_91061896610046
// MI455X (gfx1250) — compile-verified
//
#include <hip/hip_runtime.h>

// CDNA5 / gfx1250: wave32, V_WMMA_F32_16X16X4_F32 (exact f32 matrix op).
typedef float v2f __attribute__((ext_vector_type(2)));
typedef float v8f __attribute__((ext_vector_type(8)));

// FWHT over D=128 via two WMMA passes per 16x16 tile (= 2 data rows):
//   Y = blockdiag(H8,H8) * M * H16, scaled by 1/sqrt(128).
// Pass1: D1 = A(M^T) x B1(blockdiag(H8,H8)*scale)
// Pass2: D2 = A(D1-regs == D1^T*Pi) x B2(Pi^T*H16)  ==> D2 = scale*Y (D layout)
// One wave handles one 16x16 tile per iteration; wave-uniform scalar loop.
__global__ __launch_bounds__(256) void fwht128_wmma_kernel(
    const float* __restrict__ src, float* __restrict__ dst, int tiles) {
  const int lane = threadIdx.x & 31;
  const int hi   = lane >> 4;   // lane half: 0 or 1
  const int col  = lane & 15;   // N / column index

  const float scale = 0.08838834764831843f; // 1/sqrt(128)

  // Lane-resident constant B operands (loop-invariant).
  // B layout for 4x16 slice s, VGPR v, lane: element (k = 4s+v+2*hi, n = col).
  v2f b1[4], b2[4];
#pragma unroll
  for (int s = 0; s < 4; ++s) {
#pragma unroll
    for (int v = 0; v < 2; ++v) {
      const int r = 4 * s + v + 2 * hi;  // row index 0..15
      // B1 = blockdiag(H8, H8) * scale
      float e1 = 0.0f;
      if ((r >> 3) == (col >> 3))
        e1 = (__popc((r & 7) & (col & 7)) & 1) ? -scale : scale;
      // B2[k, col] = H16[pi(k), col],  pi(k) = 2*(k>>2) + (k&1) + 8*((k>>1)&1)
      const int pk = ((r >> 2) << 1) | (r & 1) | (((r >> 1) & 1) << 3);
      const float e2 = (__popc(pk & col) & 1) ? -1.0f : 1.0f;
      if (v == 0) { b1[s].x = e1; b2[s].x = e2; }
      else        { b1[s].y = e1; b2[s].y = e2; }
    }
  }

  // Scalarize the per-wave tile index so loop control is SALU and global
  // accesses use the SGPR-base + VGPR-offset addressing mode.
  const int wid = __builtin_amdgcn_readfirstlane(
      (int)blockIdx.x * ((int)blockDim.x >> 5) + ((int)threadIdx.x >> 5));
  const int nwaves = (int)gridDim.x * ((int)blockDim.x >> 5);

  const int ld_off = (hi << 5) + col;   // float offset: 32*hi + col
  const int st_off = (hi << 7) + col;   // float offset: 128*hi + col

  for (int t = wid; t < tiles; t += nwaves) {
    const float* sb_in  = src + ((long long)t << 8);
    float*       sb_out = dst + ((long long)t << 8);

    // Load A = M^T in A layout: slice s, VGPR v, lane -> M[4s+v+2hi, col]
    // float offset = 16*(4s+v+2hi) + col = 64s + 16v + 32hi + col
    const float* lb = sb_in + ld_off;
    v2f a0, a1, a2, a3;
    a0.x = __builtin_nontemporal_load(lb + 0);
    a0.y = __builtin_nontemporal_load(lb + 16);
    a1.x = __builtin_nontemporal_load(lb + 64);
    a1.y = __builtin_nontemporal_load(lb + 80);
    a2.x = __builtin_nontemporal_load(lb + 128);
    a2.y = __builtin_nontemporal_load(lb + 144);
    a3.x = __builtin_nontemporal_load(lb + 192);
    a3.y = __builtin_nontemporal_load(lb + 208);

    // Pass 1: K=16 via 4 chained 16x16x4 f32 WMMAs.
    v8f c = {};
    c = __builtin_amdgcn_wmma_f32_16x16x4_f32(false, a0, false, b1[0], (short)0, c, false, false);
    c = __builtin_amdgcn_wmma_f32_16x16x4_f32(false, a1, false, b1[1], (short)0, c, false, false);
    c = __builtin_amdgcn_wmma_f32_16x16x4_f32(false, a2, false, b1[2], (short)0, c, false, false);
    c = __builtin_amdgcn_wmma_f32_16x16x4_f32(false, a3, false, b1[3], (short)0, c, false, false);

    // Pass 2: reuse D-layout result registers directly as A slices (implicit
    // transpose + permutation Pi, compensated inside B2). No lane shuffles.
    v2f z0 = { c[0], c[1] };
    v2f z1 = { c[2], c[3] };
    v2f z2 = { c[4], c[5] };
    v2f z3 = { c[6], c[7] };

    v8f d = {};
    d = __builtin_amdgcn_wmma_f32_16x16x4_f32(false, z0, false, b2[0], (short)0, d, false, false);
    d = __builtin_amdgcn_wmma_f32_16x16x4_f32(false, z1, false, b2[1], (short)0, d, false, false);
    d = __builtin_amdgcn_wmma_f32_16x16x4_f32(false, z2, false, b2[2], (short)0, d, false, false);
    d = __builtin_amdgcn_wmma_f32_16x16x4_f32(false, z3, false, b2[3], (short)0, d, false, false);

    // Store D layout: VGPR g, lane -> Y[g+8*hi, col] at float offset 16g+128hi+col
    float* ob = sb_out + st_off;
#pragma unroll
    for (int g = 0; g < 8; ++g) __builtin_nontemporal_store(d[g], ob + 16 * g);
  }
}

extern "C" void kernel_launch(void* const* d_in, const int* in_sizes, int n_in,
                              void* d_out, int out_size, void* d_ws, size_t ws_size,
                              hipStream_t stream) {
  const float* q = (const float*)d_in[0];
  const float* k = (const float*)d_in[1];
  float* out = (float*)d_out;

  const int q_tiles = in_sizes[0] >> 8;  // 256 floats (2 rows of 128) per tile
  const int k_tiles = in_sizes[1] >> 8;

  auto blocks_for = [](int tiles) {
    long long need = ((long long)tiles + 7) / 8;  // 8 waves per 256-thread block
    if (need < 1) need = 1;
    if (need > 4096) need = 4096;
    return (int)need;
  };

  // Two uniform launches (q then k): no per-iteration source select, each
  // launch is ~256 MB of streaming work — more than enough to fill the GPU.
  fwht128_wmma_kernel<<<blocks_for(q_tiles), 256, 0, stream>>>(q, out, q_tiles);
  fwht128_wmma_kernel<<<blocks_for(k_tiles), 256, 0, stream>>>(
      k, out + (long long)in_sizes[0], k_tiles);
}